// Projection_13898514170502
// MI455X (gfx1250) — compile-verified
//
#include <hip/hip_runtime.h>
#include <cstdint>

// ---------------------------------------------------------------------------
// Trilinear interpolation of a [64,64,64,64] f32 volume at N points.
// Memory-bound gather kernel: one wave32 per point, lane l covers channels
// {2l, 2l+1}. Corner rows (256B contiguous) are staged into LDS with the
// CDNA5 async global->LDS engine, double-buffered across points so the wave
// computes point p while the async unit gathers point p+1.
// ---------------------------------------------------------------------------

#define AS_GLOBAL __attribute__((address_space(1)))
#define AS_LOCAL  __attribute__((address_space(3)))

typedef int v2i __attribute__((vector_size(8)));   // 64-bit payload type

#if __has_builtin(__builtin_amdgcn_global_load_async_to_lds_b64)
#define HAS_ASYNC_BUILTIN 1
#else
#define HAS_ASYNC_BUILTIN 0
#endif

__device__ __forceinline__ void async_ld_b64(const void* gptr, void* lptr) {
#if HAS_ASYNC_BUILTIN
  __builtin_amdgcn_global_load_async_to_lds_b64(
      (AS_GLOBAL v2i*)gptr, (AS_LOCAL v2i*)lptr, /*offset=*/0, /*cpol=*/0);
#else
  unsigned lds_off = (unsigned)(uintptr_t)(AS_LOCAL void*)lptr;
  unsigned long long ga = (unsigned long long)(uintptr_t)gptr;
  asm volatile("global_load_async_to_lds_b64 %0, %1, off"
               :: "v"(lds_off), "v"(ga) : "memory");
#endif
}

#if __has_builtin(__builtin_amdgcn_s_wait_asynccnt)
#define WAIT_ASYNCCNT(n) __builtin_amdgcn_s_wait_asynccnt(n)
#else
#define WAIT_ASYNCCNT(n) asm volatile("s_wait_asynccnt " #n ::: "memory")
#endif

// Volume constants (reference shapes are fixed): D=H=W=64, C=64.
// byte offset of voxel (ix,iy,iz) = ((ix*4096 + iy*64 + iz) * 64 ch * 4B)
//   -> ix<<20 | iy<<14 | iz<<8
#define PTS_PER_WAVE 8
#define WAVES_PER_BLK 8
#define PTS_PER_BLK (PTS_PER_WAVE * WAVES_PER_BLK)

__launch_bounds__(256)
__global__ void trilerp_async(const float* __restrict__ img,
                              const float* __restrict__ coords,
                              float* __restrict__ out, int N) {
  // per-wave double buffer: 8 corners x 32 lanes x float2 = 2KB per buffer
  __shared__ float2 stage[WAVES_PER_BLK][2][8 * 32];

  const int lane = threadIdx.x & 31;
  const int wave = threadIdx.x >> 5;
  const int p0 = blockIdx.x * PTS_PER_BLK + wave * PTS_PER_WAVE;

  float wgt[2][8];  // trilinear weights per double-buffer slot (registers)

  auto issue = [&](int p, int buf) {
    const float cx = coords[3 * p + 0] * 0.5f;   // 128/H = 2 -> /2
    const float cy = coords[3 * p + 1] * 0.5f;
    const float cz = coords[3 * p + 2] * 0.5f;
    const float x1 = floorf(cx), y1 = floorf(cy), z1 = floorf(cz);
    const float x2 = fminf(ceilf(cx), 63.0f);
    const float y2 = fminf(ceilf(cy), 63.0f);
    const float z2 = fminf(ceilf(cz), 63.0f);
    const float wx = cx - x1, wxc = x2 - cx;
    const float wy = cy - y1, wyc = y2 - cy;
    const float wz = cz - z1, wzc = z2 - cz;
    float* w = wgt[buf];
    // corner order: (x1,y1,z1)(x2,y1,z1)(x1,y2,z1)(x2,y2,z1) then z2 plane
    w[0] = wxc * wyc * wzc;  w[1] = wx * wyc * wzc;
    w[2] = wxc * wy  * wzc;  w[3] = wx * wy  * wzc;
    w[4] = wxc * wyc * wz;   w[5] = wx * wyc * wz;
    w[6] = wxc * wy  * wz;   w[7] = wx * wy  * wz;

    const uint32_t bx1 = (uint32_t)(int)x1 << 20;
    const uint32_t bx2 = (uint32_t)(int)x2 << 20;
    const uint32_t by1 = (uint32_t)(int)y1 << 14;
    const uint32_t by2 = (uint32_t)(int)y2 << 14;
    const uint32_t bz1 = (uint32_t)(int)z1 << 8;
    const uint32_t bz2 = (uint32_t)(int)z2 << 8;
    const uint32_t off[8] = {
        bx1 + by1 + bz1, bx2 + by1 + bz1, bx1 + by2 + bz1, bx2 + by2 + bz1,
        bx1 + by1 + bz2, bx2 + by1 + bz2, bx1 + by2 + bz2, bx2 + by2 + bz2};

    const char* gbase = (const char*)img + (uint32_t)(lane * 8);
#pragma unroll
    for (int k = 0; k < 8; ++k)
      async_ld_b64(gbase + off[k], &stage[wave][buf][k * 32 + lane]);
  };

  auto compute = [&](int p, int buf) {
    const float* w = wgt[buf];
    float ax = 0.0f, ay = 0.0f;
#pragma unroll
    for (int k = 0; k < 8; ++k) {
      const float2 q = stage[wave][buf][k * 32 + lane];
      ax = fmaf(w[k], q.x, ax);
      ay = fmaf(w[k], q.y, ay);
    }
    float2* o = (float2*)(out + (size_t)p * 64) + lane;
    *o = make_float2(ax, ay);
  };

  if (p0 < N) issue(p0, 0);

#pragma unroll
  for (int j = 0; j < PTS_PER_WAVE; ++j) {
    const int cur = j & 1;
    const int nxt = cur ^ 1;
    const bool have_next = (j + 1 < PTS_PER_WAVE) && (p0 + j + 1 < N);
    if (have_next) issue(p0 + j + 1, nxt);  // prefetch next point's corners
    if (p0 + j < N) {
      if (have_next) {
        WAIT_ASYNCCNT(8);  // newest 8 (next point) may remain outstanding
      } else {
        WAIT_ASYNCCNT(0);
      }
      asm volatile("" ::: "memory");  // keep LDS reads below the wait
      compute(p0 + j, cur);
    }
  }
}

extern "C" void kernel_launch(void* const* d_in, const int* in_sizes, int n_in,
                              void* d_out, int out_size, void* d_ws, size_t ws_size,
                              hipStream_t stream) {
  const float* img    = (const float*)d_in[0];  // [1,64,64,64,64] f32
  const float* coords = (const float*)d_in[1];  // [1,N,3] f32
  float* out = (float*)d_out;                   // [1,N,64] f32
  const int N = in_sizes[1] / 3;
  const int blocks = (N + PTS_PER_BLK - 1) / PTS_PER_BLK;
  trilerp_async<<<blocks, 256, 0, stream>>>(img, coords, out, N);
}